// LL1_63178968924376
// MI455X (gfx1250) — compile-verified
//
#include <hip/hip_runtime.h>
#include <hip/hip_bf16.h>

typedef float v2f __attribute__((ext_vector_type(2)));
typedef float v8f __attribute__((ext_vector_type(8)));

#define EPSBN 1e-5f

// ---------------------------------------------------------------------------
// Banded-GEMM stencil conv via V_WMMA_F32_16X16X4_F32.
// One wave computes a 16x16 output tile:  D[m][n] = sum_ky sum_s A[m][s]*B[s][n]
//   A[m][s] = in[y0+m+D*ky-PAD, xb+w*16-PAD+s]   (staged in LDS)
//   B[s][n] = w[ky][(s-n)/D] if (s-n)>=0, (s-n)%D==0, (s-n)/D<K else 0
// Reduction dim s is chunked by 4 (WMMA K=4).
// Optional fused epilogue: 1x1 conv + BN + tanh + elementwise multiply.
// ---------------------------------------------------------------------------
template <int K, int D, int PAD, bool DO_PW>
__global__ __launch_bounds__(256) void conv_wmma_k(
    const float* __restrict__ in, const float* __restrict__ wk,
    const float* __restrict__ bias,
    const float* w3p, const float* b3p, const float* gp, const float* bep,
    const float* mp, const float* vp, const float* __restrict__ mul,
    float* __restrict__ out, int H, int W) {
  constexpr int EXT = D * (K - 1);
  constexpr int R   = 16 + EXT;      // region rows
  constexpr int CC  = 128 + EXT;     // region cols (block covers 128 out cols)
  constexpr int RS  = CC + 1;        // padded row stride (bank spread)
  constexpr int S   = 16 + EXT;      // band length per 16-wide tile
  constexpr int NS4 = (S + 3) / 4;   // K=4 chunks

  __shared__ float sIn[R * RS + 4];       // +4: tolerate chunk tail overread
  __shared__ float sB[K * NS4 * 64];      // banded B, layout [ky][c][n][k]
  __shared__ float sW[K * K];

  const int tid = threadIdx.x;
  const int img = blockIdx.z;
  const int y0  = blockIdx.y * 16;
  const int xb  = blockIdx.x * 128;
  const size_t plane = (size_t)H * W;
  const float* src = in + (size_t)img * plane;

  if (tid < K * K) sW[tid] = wk[tid];
  __syncthreads();

  // Build banded weight matrix in LDS (zero off-band -> s>=S chunks are 0).
  for (int i = tid; i < K * NS4 * 64; i += 256) {
    int k  = i & 3;
    int n  = (i >> 2) & 15;
    int r  = i >> 6;
    int c  = r % NS4;
    int ky = r / NS4;
    int s  = 4 * c + k;
    int t  = s - n;
    float v = 0.f;
    if (s < S && t >= 0 && (t % D) == 0 && (t / D) < K) v = sW[ky * K + t / D];
    sB[i] = v;
  }

  // Stage input region (with zero padding, incl. LDS pad column + tail).
  for (int i = tid; i < R * RS + 4; i += 256) {
    int r = i / RS, c = i % RS;
    float v = 0.f;
    if (r < R && c < CC) {
      int gy = y0 - PAD + r;
      int gx = xb - PAD + c;
      if (gy >= 0 && gy < H && gx >= 0 && gx < W) v = src[(size_t)gy * W + gx];
    }
    sIn[i] = v;
  }
  __syncthreads();

  const int lane = tid & 31;
  const int wv   = tid >> 5;
  const int x0w  = xb + wv * 16;
  if (x0w >= W) return;  // wave-uniform: EXEC stays all-ones for WMMA

  const int Mn = lane & 15;           // A row / B col / D col for this lane
  const int kk = (lane >> 4) << 1;    // lanes 0-15 -> k 0,1 ; lanes 16-31 -> k 2,3

  v8f acc = {0.f, 0.f, 0.f, 0.f, 0.f, 0.f, 0.f, 0.f};
#pragma unroll
  for (int ky = 0; ky < K; ++ky) {
    const float* arow = &sIn[(Mn + D * ky) * RS + wv * 16];
    const float* brow = &sB[(ky * NS4) * 64 + Mn * 4];
#pragma unroll
    for (int c = 0; c < NS4; ++c) {
      v2f a, b;
      a.x = arow[4 * c + kk];
      a.y = arow[4 * c + kk + 1];
      b.x = brow[c * 64 + kk];
      b.y = brow[c * 64 + kk + 1];
      acc = __builtin_amdgcn_wmma_f32_16x16x4_f32(false, a, false, b,
                                                  (short)0, acc, false, false);
    }
  }

  const float bb = bias[0];
  float w3 = 0.f, b3 = 0.f, sc = 0.f, mm = 0.f, be = 0.f;
  if constexpr (DO_PW) {
    w3 = w3p[0];
    b3 = b3p[0];
    sc = gp[0] * rsqrtf(vp[0] + EPSBN);
    mm = mp[0];
    be = bep[0];
  }

#pragma unroll
  for (int r2 = 0; r2 < 8; ++r2) {
    int Mrow = r2 + ((lane >> 4) << 3);  // D layout: VGPR r -> M=r (lo) / r+8 (hi)
    size_t oidx = (size_t)img * plane + (size_t)(y0 + Mrow) * W + x0w + Mn;
    float z = acc[r2] + bb;
    if constexpr (DO_PW) {
      z = w3 * z + b3;
      z = (z - mm) * sc + be;
      z = tanhf(z);
      z *= mul[oidx];
    }
    out[oidx] = z;
  }
}

// ---------------------------------------------------------------------------
// 4x4 average pooling, NCHW with C=1, N=16.
// ---------------------------------------------------------------------------
__global__ __launch_bounds__(256) void pool4_k(const float* __restrict__ in,
                                               float* __restrict__ out, int H,
                                               int W) {
  int oH = H >> 2, oW = W >> 2;
  int idx = blockIdx.x * 256 + threadIdx.x;
  int total = 16 * oH * oW;
  if (idx >= total) return;
  int ox = idx % oW;
  int oy = (idx / oW) % oH;
  int n  = idx / (oW * oH);
  const float* p = in + (size_t)n * H * W + (size_t)(oy * 4) * W + ox * 4;
  float s = 0.f;
#pragma unroll
  for (int dy = 0; dy < 4; ++dy)
#pragma unroll
    for (int dx = 0; dx < 4; ++dx) s += p[dy * W + dx];
  out[idx] = s * (1.f / 16.f);
}

// ---------------------------------------------------------------------------
// Final 16x16 stage: conv3x3+BN+tanh, conv3x3+BN+tanh, mean, sigmoid.
// One block (256 threads = 16x16 pixels) per image.
// ---------------------------------------------------------------------------
struct C4P {
  const float *w1, *b1, *w2, *b2, *g1, *be1, *m1, *v1, *g2, *be2, *m2, *v2;
};

__global__ __launch_bounds__(256) void final_k(const float* __restrict__ in,
                                               C4P p, float* __restrict__ out) {
  __shared__ float t1[256];
  __shared__ float t2[256];
  __shared__ float red[256];
  __shared__ float w1s[9], w2s[9];

  int tid = threadIdx.x, img = blockIdx.x;
  int x = tid & 15, y = tid >> 4;
  t1[tid] = in[img * 256 + tid];
  if (tid < 9) { w1s[tid] = p.w1[tid]; w2s[tid] = p.w2[tid]; }
  __syncthreads();

  float s = p.b1[0];
#pragma unroll
  for (int ky = 0; ky < 3; ++ky)
#pragma unroll
    for (int kx = 0; kx < 3; ++kx) {
      int yy = y + ky - 1, xx = x + kx - 1;
      if (yy >= 0 && yy < 16 && xx >= 0 && xx < 16)
        s += w1s[ky * 3 + kx] * t1[yy * 16 + xx];
    }
  float sc1 = p.g1[0] * rsqrtf(p.v1[0] + EPSBN);
  s = tanhf((s - p.m1[0]) * sc1 + p.be1[0]);
  t2[tid] = s;
  __syncthreads();

  float s2 = p.b2[0];
#pragma unroll
  for (int ky = 0; ky < 3; ++ky)
#pragma unroll
    for (int kx = 0; kx < 3; ++kx) {
      int yy = y + ky - 1, xx = x + kx - 1;
      if (yy >= 0 && yy < 16 && xx >= 0 && xx < 16)
        s2 += w2s[ky * 3 + kx] * t2[yy * 16 + xx];
    }
  float sc2 = p.g2[0] * rsqrtf(p.v2[0] + EPSBN);
  s2 = tanhf((s2 - p.m2[0]) * sc2 + p.be2[0]);
  red[tid] = s2;
  __syncthreads();
  for (int st = 128; st > 0; st >>= 1) {
    if (tid < st) red[tid] += red[tid + st];
    __syncthreads();
  }
  if (tid == 0) {
    float m = red[0] * (1.f / 256.f);
    out[img] = 1.f / (1.f + expf(-m));
  }
}

// ---------------------------------------------------------------------------
// Host-side pipeline.
// Input flattening order (depth-first dict order):
//   0: x (16*1024*1024)
//   c1 @1, c11 @11, c2 @21, c22 @31, c3 @41, c33 @51  (w1,b1,w2,b2,w3,b3,g,be,m,v)
//   c4 @61: w1,b1,w2,b2,g1,be1,m1,v1,g2,be2,m2,v2
// ---------------------------------------------------------------------------
static inline dim3 cgrid(int H, int W) {
  return dim3((unsigned)((W + 127) / 128), (unsigned)(H / 16), 16u);
}

extern "C" void kernel_launch(void* const* d_in, const int* in_sizes, int n_in,
                              void* d_out, int out_size, void* d_ws,
                              size_t ws_size, hipStream_t stream) {
  (void)in_sizes; (void)n_in; (void)out_size; (void)ws_size;
  const float* x = (const float*)d_in[0];
  auto P = [&](int i) { return (const float*)d_in[i]; };

  float* ws = (float*)d_ws;
  size_t off = 0;
  float* U  = ws + off; off += (size_t)16 * 1024 * 1024;
  float* V  = ws + off; off += (size_t)16 * 1024 * 1024;
  float* T  = ws + off; off += (size_t)16 * 1024 * 1024;
  float* P1 = ws + off; off += (size_t)16 * 256 * 256;
  float* P2 = ws + off; off += (size_t)16 * 64 * 64;
  float* P3 = ws + off; off += (size_t)16 * 16 * 16;

  // -------- Stage 1 @ 1024x1024: c1 then c11, pool -> 256 --------
  {
    const int H = 1024, W = 1024;
    conv_wmma_k<7, 1, 3, false><<<cgrid(H, W), 256, 0, stream>>>(
        x, P(1), P(2), nullptr, nullptr, nullptr, nullptr, nullptr, nullptr,
        nullptr, U, H, W);
    conv_wmma_k<9, 4, 16, true><<<cgrid(H, W), 256, 0, stream>>>(
        U, P(3), P(4), P(5), P(6), P(7), P(8), P(9), P(10), x, T, H, W);
    conv_wmma_k<7, 1, 3, false><<<cgrid(H, W), 256, 0, stream>>>(
        T, P(11), P(12), nullptr, nullptr, nullptr, nullptr, nullptr, nullptr,
        nullptr, U, H, W);
    conv_wmma_k<9, 4, 16, true><<<cgrid(H, W), 256, 0, stream>>>(
        U, P(13), P(14), P(15), P(16), P(17), P(18), P(19), P(20), T, V, H, W);
    int total = 16 * 256 * 256;
    pool4_k<<<(total + 255) / 256, 256, 0, stream>>>(V, P1, H, W);
  }

  // -------- Stage 2 @ 256x256: c2 then c22, pool -> 64 --------
  {
    const int H = 256, W = 256;
    conv_wmma_k<5, 1, 2, false><<<cgrid(H, W), 256, 0, stream>>>(
        P1, P(21), P(22), nullptr, nullptr, nullptr, nullptr, nullptr, nullptr,
        nullptr, U, H, W);
    conv_wmma_k<7, 3, 9, true><<<cgrid(H, W), 256, 0, stream>>>(
        U, P(23), P(24), P(25), P(26), P(27), P(28), P(29), P(30), P1, T, H, W);
    conv_wmma_k<5, 1, 2, false><<<cgrid(H, W), 256, 0, stream>>>(
        T, P(31), P(32), nullptr, nullptr, nullptr, nullptr, nullptr, nullptr,
        nullptr, U, H, W);
    conv_wmma_k<7, 3, 9, true><<<cgrid(H, W), 256, 0, stream>>>(
        U, P(33), P(34), P(35), P(36), P(37), P(38), P(39), P(40), T, V, H, W);
    int total = 16 * 64 * 64;
    pool4_k<<<(total + 255) / 256, 256, 0, stream>>>(V, P2, H, W);
  }

  // -------- Stage 3 @ 64x64: c3 (x2) then c33 (also on x2!), pool -> 16 -----
  {
    const int H = 64, W = 64;
    conv_wmma_k<5, 1, 2, false><<<cgrid(H, W), 256, 0, stream>>>(
        P2, P(41), P(42), nullptr, nullptr, nullptr, nullptr, nullptr, nullptr,
        nullptr, U, H, W);
    // x3a = branch(x2, c3) * x2
    conv_wmma_k<5, 3, 6, true><<<cgrid(H, W), 256, 0, stream>>>(
        U, P(43), P(44), P(45), P(46), P(47), P(48), P(49), P(50), P2, T, H, W);
    // reference quirk: second branch input is x2 (P2), multiplier is x3a (T)
    conv_wmma_k<5, 1, 2, false><<<cgrid(H, W), 256, 0, stream>>>(
        P2, P(51), P(52), nullptr, nullptr, nullptr, nullptr, nullptr, nullptr,
        nullptr, U, H, W);
    conv_wmma_k<5, 3, 6, true><<<cgrid(H, W), 256, 0, stream>>>(
        U, P(53), P(54), P(55), P(56), P(57), P(58), P(59), P(60), T, V, H, W);
    int total = 16 * 16 * 16;
    pool4_k<<<(total + 255) / 256, 256, 0, stream>>>(V, P3, H, W);
  }

  // -------- Stage 4 @ 16x16: conv3x3+BN+tanh x2, mean, sigmoid --------
  {
    C4P c4{P(61), P(62), P(63), P(64), P(65), P(66),
           P(67), P(68), P(69), P(70), P(71), P(72)};
    final_k<<<16, 256, 0, stream>>>(P3, c4, (float*)d_out);
  }
}